// AttentionModule_68436008894722
// MI455X (gfx1250) — compile-verified
//
#include <hip/hip_runtime.h>
#include <hip/hip_bf16.h>

typedef __attribute__((ext_vector_type(2))) float v2f;
typedef __attribute__((ext_vector_type(8))) float v8f;

#define NEG_INF (-__builtin_inff())

// ---------------------------------------------------------------------------
// K0: u = embed_w @ (key_w @ wk), wk = attn_w[64:128,0].   1 block, 128 thr.
// ---------------------------------------------------------------------------
__global__ void k0_build_u(const float* __restrict__ key_w,
                           const float* __restrict__ attn_w,
                           const float* __restrict__ embed_w,
                           float* __restrict__ u) {
  __shared__ float v[64];
  const int t = threadIdx.x;             // 0..127
  if (t < 64) {
    float s = 0.f;
    #pragma unroll 8
    for (int j = 0; j < 64; ++j) s += key_w[t * 64 + j] * attn_w[64 + j];
    v[t] = s;
  }
  __syncthreads();
  float s = 0.f;
  #pragma unroll 8
  for (int k = 0; k < 64; ++k) s += embed_w[t * 64 + k] * v[k];
  u[t] = s;
}

// ---------------------------------------------------------------------------
// K1: single streaming pass over X.
//   scores[i] = x_i . u   (16 rows/wave/tile via V_WMMA_F32_16X16X4_F32)
//   online softmax accumulation of (m, s, acc[128] = sum w~_i * x_i)
//   per-block partials -> blkrec[blk] = {m_b, s_b, y[128]}
// ---------------------------------------------------------------------------
__global__ void __launch_bounds__(256)
k1_scores_acc(const float* __restrict__ X, const float* __restrict__ u,
              float* __restrict__ scores, float* __restrict__ blkrec, int N) {
  const int lane = threadIdx.x & 31;
  const int wave = threadIdx.x >> 5;               // 0..7
  const int wavesPerBlock = blockDim.x >> 5;       // 8
  const int gwave  = blockIdx.x * wavesPerBlock + wave;
  const int nwaves = gridDim.x * wavesPerBlock;
  const int ntiles = (N + 15) >> 4;
  const int hf   = lane >> 4;                      // lane half (K split)
  const int lrow = lane & 15;                      // row within 16-row tile

  // A-matrix (16x4 f32, rows all equal to u-chunk): lane needs
  // u[4c + 2*hf], u[4c+1 + 2*hf]  -> preload all 32 chunk-pairs.
  v2f ua[32];
  #pragma unroll
  for (int c = 0; c < 32; ++c)
    ua[c] = *(const v2f*)(u + 4 * c + 2 * hf);

  float m_run = NEG_INF, s_run = 0.f;
  float a0 = 0.f, a1 = 0.f, a2 = 0.f, a3 = 0.f;    // cols 4*lane .. 4*lane+3

  for (int t = gwave; t < ntiles; t += nwaves) {
    const int base = t << 4;
    const int row  = base + lrow;
    const int rowc = row < N ? row : (N - 1);
    const float* rp = X + (size_t)rowc * 128 + 2 * hf;   // B layout K offset

    // scores for 16 rows: D += A(16x4) x B(4x16), K = 128 in 32 chunks.
    v8f d0 = {}; v8f d1 = {};
    #pragma unroll
    for (int c = 0; c < 32; c += 2) {
      v2f b0 = *(const v2f*)(rp + 4 * c);
      v2f b1 = *(const v2f*)(rp + 4 * (c + 1));
      d0 = __builtin_amdgcn_wmma_f32_16x16x4_f32(false, ua[c],     false, b0,
                                                 (short)0, d0, false, false);
      d1 = __builtin_amdgcn_wmma_f32_16x16x4_f32(false, ua[c + 1], false, b1,
                                                 (short)0, d1, false, false);
    }
    // A rows identical -> every D element in a lane = score of row (lane&15)
    float sv = d0[0] + d1[0];
    if (row >= N) sv = NEG_INF;
    if (hf == 0 && row < N) scores[row] = sv;      // stage raw scores

    // tile max (duplicated halves are harmless)
    float tmax = sv;
    #pragma unroll
    for (int off = 16; off > 0; off >>= 1)
      tmax = fmaxf(tmax, __shfl_xor(tmax, off, 32));

    if (tmax > m_run) {                            // online rescale
      const float sc = __expf(m_run - tmax);
      s_run *= sc; a0 *= sc; a1 *= sc; a2 *= sc; a3 *= sc;
      m_run = tmax;
    }
    // weighted accumulation: tile is WGP$-hot, re-read column-partitioned
    #pragma unroll
    for (int m = 0; m < 16; ++m) {
      const float sm = __shfl(sv, m, 32);
      const float w  = __expf(sm - m_run);         // 0 for padded rows
      int r = base + m; if (r >= N) r = N - 1;
      const float4 x4 = *(const float4*)(X + (size_t)r * 128 + lane * 4);
      s_run += w;
      a0 += w * x4.x; a1 += w * x4.y; a2 += w * x4.z; a3 += w * x4.w;
    }
  }

  // combine the block's 8 waves
  __shared__ float l_m[8], l_s[8];
  __shared__ float l_acc[8][128];
  l_acc[wave][lane * 4 + 0] = a0;
  l_acc[wave][lane * 4 + 1] = a1;
  l_acc[wave][lane * 4 + 2] = a2;
  l_acc[wave][lane * 4 + 3] = a3;
  if (lane == 0) { l_m[wave] = m_run; l_s[wave] = s_run; }
  __syncthreads();
  if (threadIdx.x < 128) {
    const int p = threadIdx.x;
    float mb = NEG_INF;
    for (int w = 0; w < 8; ++w) mb = fmaxf(mb, l_m[w]);
    float y = 0.f, sb = 0.f;
    for (int w = 0; w < 8; ++w) {
      const float f = (l_m[w] == NEG_INF) ? 0.f : __expf(l_m[w] - mb);
      y  += f * l_acc[w][p];
      sb += f * l_s[w];
    }
    float* rec = blkrec + (size_t)blockIdx.x * 130;
    rec[2 + p] = y;
    if (p == 0) { rec[0] = mb; rec[1] = sb; }
  }
}

// ---------------------------------------------------------------------------
// K2: combine block partials -> M, Z, Y[128]; out[0:64] = (Y/Z)@W_e + b_e.
// 1 block, 128 threads.
// ---------------------------------------------------------------------------
__global__ void k2_combine(const float* __restrict__ blkrec, int NB,
                           const float* __restrict__ embed_w,
                           const float* __restrict__ embed_b,
                           float* __restrict__ out, float* __restrict__ hdr) {
  __shared__ float red[128];
  __shared__ float Y[128];
  const int tid = threadIdx.x;
  float m = NEG_INF;
  for (int b = tid; b < NB; b += 128) m = fmaxf(m, blkrec[(size_t)b * 130]);
  red[tid] = m; __syncthreads();
  for (int s = 64; s > 0; s >>= 1) {
    if (tid < s) red[tid] = fmaxf(red[tid], red[tid + s]);
    __syncthreads();
  }
  const float M = red[0];
  __syncthreads();
  float z = 0.f;
  for (int b = tid; b < NB; b += 128) {
    const float mb = blkrec[(size_t)b * 130];
    const float f  = (mb == NEG_INF) ? 0.f : __expf(mb - M);
    z += f * blkrec[(size_t)b * 130 + 1];
  }
  red[tid] = z; __syncthreads();
  for (int s = 64; s > 0; s >>= 1) {
    if (tid < s) red[tid] += red[tid + s];
    __syncthreads();
  }
  const float Z = red[0];
  float y = 0.f;
  for (int b = 0; b < NB; ++b) {
    const float mb = blkrec[(size_t)b * 130];
    const float f  = (mb == NEG_INF) ? 0.f : __expf(mb - M);
    y += f * blkrec[(size_t)b * 130 + 2 + tid];
  }
  Y[tid] = y / Z;
  if (tid == 0) { hdr[0] = M; hdr[1] = Z; }
  __syncthreads();
  if (tid < 64) {
    float o = embed_b[tid];
    #pragma unroll 8
    for (int p = 0; p < 128; ++p) o += Y[p] * embed_w[(size_t)p * 64 + tid];
    out[tid] = o;
  }
}

// ---------------------------------------------------------------------------
// K3: in-place normalize staged scores -> attention weights.
// ---------------------------------------------------------------------------
__global__ void k3_normalize(float* __restrict__ wout,
                             const float* __restrict__ hdr, int N) {
  const int i = blockIdx.x * blockDim.x + threadIdx.x;
  if (i < N) {
    const float M = hdr[0];
    const float invZ = 1.0f / hdr[1];
    wout[i] = __expf(wout[i] - M) * invZ;
  }
}

// ---------------------------------------------------------------------------
extern "C" void kernel_launch(void* const* d_in, const int* in_sizes, int n_in,
                              void* d_out, int out_size, void* d_ws,
                              size_t ws_size, hipStream_t stream) {
  const float* X       = (const float*)d_in[1];  // neighbor_states (N,128)
  const float* embed_w = (const float*)d_in[2];  // (128,64)
  const float* embed_b = (const float*)d_in[3];  // (64,)
  const float* key_w   = (const float*)d_in[4];  // (64,64)
  const float* attn_w  = (const float*)d_in[8];  // (128,1)
  // my_state, key_b, query_w/b, attn_b drop out (softmax shift invariance).

  const int N = in_sizes[1] / 128;

  float* out_emb = (float*)d_out;        // weighted_embedding (64)
  float* out_w   = (float*)d_out + 64;   // attention_weights (N); scores staged here

  float* ws     = (float*)d_ws;
  float* u      = ws;                    // 128
  float* hdr    = ws + 128;              // M, Z
  float* blkrec = ws + 192;              // NB * 130

  const int NB = 512;                    // 512 blocks x 8 waves = 4096 waves

  k0_build_u<<<1, 128, 0, stream>>>(key_w, attn_w, embed_w, u);
  k1_scores_acc<<<NB, 256, 0, stream>>>(X, u, out_w, blkrec, N);
  k2_combine<<<1, 128, 0, stream>>>(blkrec, NB, embed_w, embed_b, out_emb, hdr);
  k3_normalize<<<(N + 255) / 256, 256, 0, stream>>>(out_w, hdr, N);
}